// MultiBoxLoss_60662118089050
// MI455X (gfx1250) — compile-verified
//
#include <hip/hip_runtime.h>
#include <cstdint>
#include <cstddef>

#define NB 128
#define NP 8732
#define NC 21
#define NT 16

typedef unsigned int u32x4 __attribute__((ext_vector_type(4)));
typedef int          i32x8 __attribute__((ext_vector_type(8)));
typedef int          i32x4 __attribute__((ext_vector_type(4)));

static __device__ __forceinline__ unsigned ballot32(bool b) {
#if defined(__has_builtin)
#if __has_builtin(__builtin_amdgcn_ballot_w32)
  return __builtin_amdgcn_ballot_w32(b);
#else
  return (unsigned)__ballot(b);
#endif
#else
  return (unsigned)__ballot(b);
#endif
}

static __device__ __forceinline__ void wait_tensorcnt0() {
#if defined(__has_builtin)
#if __has_builtin(__builtin_amdgcn_s_wait_tensorcnt)
  __builtin_amdgcn_s_wait_tensorcnt(0);
#else
  asm volatile("s_wait_tensorcnt 0x0" ::: "memory");
#endif
#else
  asm volatile("s_wait_tensorcnt 0x0" ::: "memory");
#endif
}

// ---------------------------------------------------------------------------
// Kernel 0: zero the two loss accumulators (harness poisons d_ws).
// ---------------------------------------------------------------------------
__global__ void k_zero(float* acc) {
  if (threadIdx.x == 0) { acc[0] = 0.0f; acc[1] = 0.0f; }
}

// ---------------------------------------------------------------------------
// Kernel 1: per-batch matching (one workgroup per batch row).
//  - IoU of 16 truths vs 8732 priors, best-truth per prior kept in LDS,
//    best-prior per truth via packed ds_max_u64 atomics (ties -> smallest p).
//  - Sequential best_prior override (2.0 / truth index, last truth wins).
//  - conf_t (u8) written, smooth-L1 loc loss over positives accumulated.
// ---------------------------------------------------------------------------
__global__ __launch_bounds__(256) void k_match(
    const float* __restrict__ loc, const float* __restrict__ priors,
    const float* __restrict__ truths, const int* __restrict__ labels,
    unsigned char* __restrict__ conf_t, int* __restrict__ num_pos,
    float* __restrict__ acc)
{
  __shared__ float s_tr[NT][4];
  __shared__ float s_area[NT];
  __shared__ int   s_lb[NT];
  __shared__ unsigned long long s_bestp[NT];
  __shared__ float s_bto[NP];              // best_truth_overlap
  __shared__ unsigned char s_bti[NP];      // best_truth_idx
  __shared__ float s_rs[256];
  __shared__ int   s_rc[256];

  const int b = blockIdx.x;
  const int tid = threadIdx.x;

  if (tid < NT) {
    const float* tr = truths + ((size_t)b * NT + tid) * 4;
    float x1 = tr[0], y1 = tr[1], x2 = tr[2], y2 = tr[3];
    s_tr[tid][0] = x1; s_tr[tid][1] = y1; s_tr[tid][2] = x2; s_tr[tid][3] = y2;
    s_area[tid] = (x2 - x1) * (y2 - y1);
    s_lb[tid] = labels[(size_t)b * NT + tid];
    s_bestp[tid] = 0ULL;
  }
  __syncthreads();

  const float4* pr4 = reinterpret_cast<const float4*>(priors);

  float lbo[NT]; int lbp[NT];
#pragma unroll
  for (int t = 0; t < NT; ++t) { lbo[t] = -1.0f; lbp[t] = 0; }

  for (int p = tid; p < NP; p += 256) {
    float4 q = pr4[p];                       // cx, cy, w, h
    float bx1 = q.x - q.z * 0.5f, by1 = q.y - q.w * 0.5f;
    float bx2 = q.x + q.z * 0.5f, by2 = q.y + q.w * 0.5f;
    float area_b = q.z * q.w;
    float bo = -1.0f; int bt = 0;
#pragma unroll
    for (int t = 0; t < NT; ++t) {
      float lx = fmaxf(s_tr[t][0], bx1), ly = fmaxf(s_tr[t][1], by1);
      float rx = fminf(s_tr[t][2], bx2), ry = fminf(s_tr[t][3], by2);
      float w = fmaxf(rx - lx, 0.0f), h = fmaxf(ry - ly, 0.0f);
      float inter = w * h;
      float iou = inter / (s_area[t] + area_b - inter);
      if (iou > bo) { bo = iou; bt = t; }             // first max wins (t asc)
      if (iou > lbo[t]) { lbo[t] = iou; lbp[t] = p; } // first max wins (p asc)
    }
    s_bto[p] = bo; s_bti[p] = (unsigned char)bt;
  }
#pragma unroll
  for (int t = 0; t < NT; ++t) {
    // pack (iou_bits, NP - p): max -> highest iou, ties -> smallest p
    unsigned long long key =
        ((unsigned long long)__float_as_uint(lbo[t]) << 32) |
        (unsigned)(NP - lbp[t]);
    atomicMax(&s_bestp[t], key);
  }
  __syncthreads();

  if (tid == 0) {
    for (int t = 0; t < NT; ++t) {   // sequential: later t wins on conflicts
      int p = NP - (int)(s_bestp[t] & 0xFFFFFFFFULL);
      s_bto[p] = 2.0f;
      s_bti[p] = (unsigned char)t;
    }
  }
  __syncthreads();

  const float4* loc4 = reinterpret_cast<const float4*>(loc);
  float lsum = 0.0f; int cnt = 0;
  for (int p = tid; p < NP; p += 256) {
    int t = s_bti[p];
    int conf = (s_bto[p] < 0.5f) ? 0 : (s_lb[t] + 1);
    conf_t[(size_t)b * NP + p] = (unsigned char)conf;
    if (conf > 0) {
      ++cnt;
      float4 q = pr4[p];
      float mx1 = s_tr[t][0], my1 = s_tr[t][1], mx2 = s_tr[t][2], my2 = s_tr[t][3];
      float g0 = ((mx1 + mx2) * 0.5f - q.x) / (0.1f * q.z);
      float g1 = ((my1 + my2) * 0.5f - q.y) / (0.1f * q.w);
      float g2 = logf((mx2 - mx1) / q.z) / 0.2f;
      float g3 = logf((my2 - my1) / q.w) / 0.2f;
      float4 l = loc4[(size_t)b * NP + p];
      float d0 = l.x - g0, d1 = l.y - g1, d2 = l.z - g2, d3 = l.w - g3;
      float a0 = fabsf(d0), a1 = fabsf(d1), a2 = fabsf(d2), a3 = fabsf(d3);
      lsum += (a0 < 1.0f ? 0.5f * d0 * d0 : a0 - 0.5f);
      lsum += (a1 < 1.0f ? 0.5f * d1 * d1 : a1 - 0.5f);
      lsum += (a2 < 1.0f ? 0.5f * d2 * d2 : a2 - 0.5f);
      lsum += (a3 < 1.0f ? 0.5f * d3 * d3 : a3 - 0.5f);
    }
  }
  s_rs[tid] = lsum; s_rc[tid] = cnt;
  __syncthreads();
  for (int s = 128; s > 0; s >>= 1) {
    if (tid < s) { s_rs[tid] += s_rs[tid + s]; s_rc[tid] += s_rc[tid + s]; }
    __syncthreads();
  }
  if (tid == 0) { atomicAdd(&acc[0], s_rs[0]); num_pos[b] = s_rc[0]; }
}

// ---------------------------------------------------------------------------
// Kernel 2: per-prior CE loss. A 256-prior tile of conf_data (21 KB, fully
// contiguous, 16B aligned) is DMA'd into LDS by the Tensor Data Mover
// (tensor_load_to_lds, TENSORcnt; 6-arg clang-23 builtin form); fallback is
// per-lane async global->LDS b128 copies (ASYNCcnt). Each thread then does a
// 21-wide logsumexp from LDS (stride 21 floats -> bank-conflict-free on
// wave32).
// ---------------------------------------------------------------------------
__global__ __launch_bounds__(256) void k_ce(
    const float* __restrict__ conf, const unsigned char* __restrict__ conf_t,
    float* __restrict__ lca)
{
  __shared__ __align__(16) float s_tile[256 * NC];

  const int b  = blockIdx.y;
  const int p0 = blockIdx.x * 256;
  const int nt = (NP - p0 < 256) ? (NP - p0) : 256;   // 256 or 28, both %4==0
  const int tid = threadIdx.x;

  const float* gsrc = conf + ((size_t)b * NP + p0) * NC;  // 16B aligned
  unsigned lds0 = (unsigned)(uintptr_t)s_tile;

#if defined(__has_builtin) && __has_builtin(__builtin_amdgcn_tensor_load_to_lds)
  if (tid < 32) {                 // one wave issues the tile DMA (EXEC ignored)
    unsigned long long ga = (unsigned long long)(uintptr_t)gsrc;
    unsigned nelem = (unsigned)(nt * NC);               // <= 5376 elements
    u32x4 g0;
    g0.x = 1u;                                          // count=1 descriptor
    g0.y = lds0;                                        // lds_addr (bytes)
    g0.z = (unsigned)(ga & 0xFFFFFFFFull);              // global_addr[31:0]
    g0.w = (unsigned)((ga >> 32) & 0x01FFFFFFull) | 0x80000000u; // [56:32]|type=2
    i32x8 g1;
    g1[0] = (int)(2u << 16);                            // data_size = 4 bytes
    g1[1] = (int)((nelem & 0xFFFFu) << 16);             // tensor_dim0[15:0]
    g1[2] = (int)(((nelem >> 16) & 0xFFFFu) | (1u << 16)); // td0[31:16]|td1=1
    g1[3] = (int)(nelem << 16);                         // tile_dim0 = nelem
    g1[4] = 0;                                          // tile_dim1/2 unused
    g1[5] = (int)nelem;                                 // td0_stride[31:0]
    g1[6] = (int)((nelem & 0xFFFFu) << 16);             // td1_stride[15:0]
    g1[7] = (int)(nelem >> 16);                         // td1_stride[47:16]
    i32x4 g2 = {0, 0, 0, 0};
    i32x4 g3 = {0, 0, 0, 0};
    i32x8 g4 = {0, 0, 0, 0, 0, 0, 0, 0};                // unused trailing group
    __builtin_amdgcn_tensor_load_to_lds(g0, g1, g2, g3, g4, 0);
    wait_tensorcnt0();
  }
  __syncthreads();
#else
  {
    const int n128 = (nt * NC) / 4;                     // # of 16B chunks
    for (int i = tid; i < n128; i += 256) {
      unsigned ldsa = lds0 + (unsigned)i * 16u;
      unsigned voff = (unsigned)i * 16u;
      asm volatile("global_load_async_to_lds_b128 %0, %1, %2"
                   :: "v"(ldsa), "v"(voff), "s"(gsrc) : "memory");
    }
    asm volatile("s_wait_asynccnt 0x0" ::: "memory");
    __syncthreads();
  }
#endif

  if (tid < nt) {
    const float* r = &s_tile[tid * NC];
    float m = r[0];
#pragma unroll
    for (int c = 1; c < NC; ++c) m = fmaxf(m, r[c]);
    float s = 0.0f;
#pragma unroll
    for (int c = 0; c < NC; ++c) s += expf(r[c] - m);
    float lse = m + logf(s);
    int p = p0 + tid;
    int ct = conf_t[(size_t)b * NP + p];
    lca[(size_t)b * NP + p] = lse - r[ct];
  }
}

// ---------------------------------------------------------------------------
// Kernel 3: hard negative mining + classification loss (one block per row).
// The 35 KB loss row is staged into LDS with async global->LDS b128 copies.
// mined losses are >= 0 so float bits order as u32: LDS radix-select (4 byte
// passes) finds the num_neg-th largest; exact stable-sort tie-break by index
// is reproduced with a wave32 ballot-based ordered prefix scan.
// ---------------------------------------------------------------------------
__global__ __launch_bounds__(1024) void k_mine(
    const float* __restrict__ lca, const unsigned char* __restrict__ conf_t,
    const int* __restrict__ num_pos, float* __restrict__ acc)
{
  __shared__ unsigned s_bits[NP];
  __shared__ __align__(16) float s_lca[NP];
  __shared__ unsigned char s_pos[NP];
  __shared__ unsigned s_hist[256];
  __shared__ unsigned s_sel[2];           // [0]=prefix/thresh, [1]=remaining k
  __shared__ unsigned s_wsum[32], s_wpre[32];
  __shared__ unsigned s_base, s_ctot;
  __shared__ float    s_red[1024];

  const int b = blockIdx.x;
  const int tid = threadIdx.x;

  // Async-stage the whole loss row (34928 B, 16B aligned) into s_lca.
  {
    const float* grow = lca + (size_t)b * NP;
    unsigned lds0 = (unsigned)(uintptr_t)s_lca;
    const int n128 = (NP * 4) / 16;                     // 2183 chunks
    for (int i = tid; i < n128; i += 1024) {
      unsigned ldsa = lds0 + (unsigned)i * 16u;
      unsigned voff = (unsigned)i * 16u;
      asm volatile("global_load_async_to_lds_b128 %0, %1, %2"
                   :: "v"(ldsa), "v"(voff), "s"(grow) : "memory");
    }
    asm volatile("s_wait_asynccnt 0x0" ::: "memory");
    __syncthreads();
  }

  for (int p = tid; p < NP; p += 1024) {
    bool pos = conf_t[(size_t)b * NP + p] > 0;
    s_pos[p] = pos ? 1 : 0;
    s_bits[p] = pos ? 0u : __float_as_uint(s_lca[p]);  // mined = where(pos,0,loss)
  }
  __syncthreads();

  int k = 3 * num_pos[b];
  if (k > NP - 1) k = NP - 1;                    // jnp.clip(a_max=P-1)

  unsigned thresh, kfin;
  if (k > 0) {                                   // uniform branch per block
    if (tid == 0) { s_sel[0] = 0u; s_sel[1] = (unsigned)k; }
    __syncthreads();
    for (int pass = 3; pass >= 0; --pass) {
      if (tid < 256) s_hist[tid] = 0u;
      __syncthreads();
      unsigned pre = s_sel[0];
      unsigned hmask = (pass == 3) ? 0u : (0xFFFFFFFFu << ((pass + 1) * 8));
      for (int p = tid; p < NP; p += 1024) {
        unsigned u = s_bits[p];
        if ((u & hmask) == (pre & hmask))
          atomicAdd(&s_hist[(u >> (pass * 8)) & 255u], 1u);
      }
      __syncthreads();
      if (tid == 0) {
        unsigned remk = s_sel[1], a = 0; int v = 255;
        for (; v > 0; --v) {
          unsigned c = s_hist[v];
          if (a + c >= remk) break;
          a += c;
        }
        s_sel[0] = pre | ((unsigned)v << (pass * 8));
        s_sel[1] = remk - a;
      }
      __syncthreads();
    }
    thresh = s_sel[0]; kfin = s_sel[1];
  } else {
    thresh = 0xFFFFFFFFu; kfin = 0u;
  }

  if (tid == 0) s_base = 0u;
  __syncthreads();

  float fsum = 0.0f;
  for (int c0 = 0; c0 < NP; c0 += 1024) {
    int p = c0 + tid;
    bool valid = p < NP;
    unsigned u = valid ? s_bits[p] : 0u;
    bool eq = valid && (u == thresh);
    unsigned m = ballot32(eq);
    int lane = tid & 31, wid = tid >> 5;
    if (lane == 0) s_wsum[wid] = (unsigned)__popc(m);
    __syncthreads();
    if (tid == 0) {
      unsigned a = 0;
      for (int w = 0; w < 32; ++w) { s_wpre[w] = a; a += s_wsum[w]; }
      s_ctot = a;
    }
    __syncthreads();
    unsigned eqrank = s_base + s_wpre[wid] +
                      (unsigned)__popc(m & ((1u << lane) - 1u));
    bool sel = valid && (s_pos[p] || (u > thresh) || (eq && eqrank < kfin));
    if (sel) fsum += s_lca[p];
    __syncthreads();
    if (tid == 0) s_base += s_ctot;
    __syncthreads();
  }

  s_red[tid] = fsum;
  __syncthreads();
  for (int s = 512; s > 0; s >>= 1) {
    if (tid < s) s_red[tid] += s_red[tid + s];
    __syncthreads();
  }
  if (tid == 0) atomicAdd(&acc[1], s_red[0]);
}

// ---------------------------------------------------------------------------
// Kernel 4: finalize. N = sum(num_pos); out = (loss_l/N, loss_c/N).
// ---------------------------------------------------------------------------
__global__ void k_final(const float* acc, const int* npos, float* out) {
  if (threadIdx.x == 0 && blockIdx.x == 0) {
    int n = 0;
    for (int i = 0; i < NB; ++i) n += npos[i];
    float N = (float)n;
    out[0] = acc[0] / N;
    out[1] = acc[1] / N;
  }
}

// ---------------------------------------------------------------------------
extern "C" void kernel_launch(void* const* d_in, const int* in_sizes, int n_in,
                              void* d_out, int out_size, void* d_ws, size_t ws_size,
                              hipStream_t stream) {
  (void)in_sizes; (void)n_in; (void)out_size; (void)ws_size;
  const float* loc    = (const float*)d_in[0];   // [B,P,4]
  const float* conf   = (const float*)d_in[1];   // [B,P,C]
  const float* priors = (const float*)d_in[2];   // [P,4]
  const float* truths = (const float*)d_in[3];   // [B,T,4]
  const int*   labels = (const int*)d_in[4];     // [B,T]
  float* out = (float*)d_out;

  char* ws = (char*)d_ws;
  float* acc = (float*)ws;                                   // 2 floats
  int*   npos = (int*)(ws + 16);                             // NB ints
  unsigned char* conf_t = (unsigned char*)(ws + 1024);       // NB*NP u8
  float* lca = (float*)(ws + 1024 + (size_t)NB * NP);        // NB*NP f32, 16B aligned

  k_zero<<<1, 1, 0, stream>>>(acc);
  k_match<<<NB, 256, 0, stream>>>(loc, priors, truths, labels, conf_t, npos, acc);
  dim3 g2((NP + 255) / 256, NB);
  k_ce<<<g2, 256, 0, stream>>>(conf, conf_t, lca);
  k_mine<<<NB, 1024, 0, stream>>>(lca, conf_t, npos, acc);
  k_final<<<1, 1, 0, stream>>>(acc, npos, out);
}